// VOSHead_41850161332614
// MI455X (gfx1250) — compile-verified
//
#include <hip/hip_runtime.h>
#include <math.h>

typedef __attribute__((ext_vector_type(2))) float v2f;
typedef __attribute__((ext_vector_type(8))) float v8f;

#define C_ 10
#define Q_ 1000
#define R_ 1024
#define N_ 512
#define S_ 10000
#define ROWS_TOT 10000
#define ROWS_PAD 10048   /* 157 * 64 */
#define KCHUNKS 157

__device__ __forceinline__ const float* resolve_row(int code, const float* idd,
                                                    const float* boxf, const float* zrow){
  if (code < 0) return zrow;
  if (code < ROWS_TOT) return idd + (size_t)code * R_;
  return boxf + (size_t)(code - ROWS_TOT) * R_;
}

// ---------------------------------------------------------------------------
// K0: build gather-index table for post-update queues, zero helpers.
// Queues are full (counter==Q): update = roll-by-k_c then append class samples.
// idx[g] in [0,10000): old id_data row; [10000,10512): box_features row-10000;
// -1: zero row (K padding).
// ---------------------------------------------------------------------------
__global__ void k_init(const int* __restrict__ labels, int* __restrict__ idxArr,
                       float* __restrict__ zrow, unsigned long long* __restrict__ best){
  __shared__ int cnt[C_];
  const int tid = threadIdx.x;
  for (int i = tid; i < R_; i += 256) zrow[i] = 0.f;
  if (tid < C_){ best[tid] = 0ull; cnt[tid] = 0; }
  __syncthreads();
  if (tid == 0){
    for (int n = 0; n < N_; ++n) ++cnt[labels[n]];
  }
  __syncthreads();
  for (int g = tid; g < ROWS_TOT; g += 256){
    const int c = g / Q_, q = g - c*Q_;
    const int k = cnt[c];
    if (q < Q_ - k) idxArr[g] = g + k;           // shifted old rows
  }
  if (tid == 0){                                  // appended rows, arrival order
    int pos[C_];
    for (int c = 0; c < C_; ++c) pos[c] = c*Q_ + (Q_ - cnt[c]);
    for (int n = 0; n < N_; ++n){
      const int c = labels[n];
      idxArr[pos[c]++] = ROWS_TOT + n;
    }
  }
  for (int g = ROWS_TOT + tid; g < ROWS_PAD; g += 256) idxArr[g] = -1;
}

// ---------------------------------------------------------------------------
// K1: scores = F @ W^T + b   [512,10]
// ---------------------------------------------------------------------------
__global__ void k_scores(const float* __restrict__ F, const float* __restrict__ W,
                         const float* __restrict__ bp, float* __restrict__ scores){
  __shared__ float red[64*C_];
  const int n = blockIdx.x, t = threadIdx.x;
  float acc[C_];
  #pragma unroll
  for (int c = 0; c < C_; ++c) acc[c] = 0.f;
  const float* f = F + (size_t)n*R_;
  for (int rr = t; rr < R_; rr += 64){
    const float fv = f[rr];
    #pragma unroll
    for (int c = 0; c < C_; ++c) acc[c] += fv * W[c*R_ + rr];
  }
  #pragma unroll
  for (int c = 0; c < C_; ++c) red[t*C_ + c] = acc[c];
  __syncthreads();
  if (t < C_){
    float s = bp[t];
    for (int j = 0; j < 64; ++j) s += red[j*C_ + t];
    scores[n*C_ + t] = s;
  }
}

// ---------------------------------------------------------------------------
// K2: sigmoid focal loss -> out[0]
// ---------------------------------------------------------------------------
__global__ void k_cls(const float* __restrict__ scores, const int* __restrict__ labels,
                      float* __restrict__ out){
  __shared__ float red[512];
  const int n = threadIdx.x;
  const int lab = labels[n];
  float s = 0.f;
  #pragma unroll
  for (int c = 0; c < C_; ++c){
    const float x  = scores[n*C_ + c];
    const float tt = (c == lab) ? 1.f : 0.f;
    const float p  = 1.f / (1.f + expf(-x));
    const float ce = fmaxf(x, 0.f) - x*tt + log1pf(expf(-fabsf(x)));
    const float pt = p*tt + (1.f - p)*(1.f - tt);
    const float at = 0.25f*tt + 0.75f*(1.f - tt);
    const float om = 1.f - pt;
    s += at * ce * om * om;
  }
  red[n] = s;
  __syncthreads();
  for (int off = 256; off > 0; off >>= 1){
    if (n < off) red[n] += red[n + off];
    __syncthreads();
  }
  if (n == 0) out[0] = red[0] / (float)N_;
}

// ---------------------------------------------------------------------------
// K3: per-class mean of updated queues (via gather indices)
// ---------------------------------------------------------------------------
__global__ void k_mu(const int* __restrict__ idxArr, const float* __restrict__ idd,
                     const float* __restrict__ F, const float* __restrict__ zrow,
                     float* __restrict__ mu){
  const int c = blockIdx.x, t = threadIdx.x;
  float a0=0.f, a1=0.f, a2=0.f, a3=0.f;
  for (int q = 0; q < Q_; ++q){
    const float* rp = resolve_row(idxArr[c*Q_ + q], idd, F, zrow);
    a0 += rp[t]; a1 += rp[t+256]; a2 += rp[t+512]; a3 += rp[t+768];
  }
  const float inv = 1.0f / (float)Q_;
  mu[c*R_ + t      ] = a0*inv;
  mu[c*R_ + t + 256] = a1*inv;
  mu[c*R_ + t + 512] = a2*inv;
  mu[c*R_ + t + 768] = a3*inv;
}

// ---------------------------------------------------------------------------
// K4: G = D^T D  (SYRK, 1024x1024x10000 f32) via V_WMMA_F32_16X16X4_F32.
// Triangular 64x64 tile grid (136 blocks), 8 waves/block:
//   waves 0-3: row strips, K rows [0,32) of staged chunk
//   waves 4-7: same row strips, K rows [32,64); halves summed via LDS.
// LDS row stride 72 avoids bank conflicts for the 16x4 fragment pattern.
// ---------------------------------------------------------------------------
__global__ void __launch_bounds__(256) k_gemm(const int* __restrict__ idxArr,
        const float* __restrict__ idd, const float* __restrict__ F,
        const float* __restrict__ zrow, float* __restrict__ G){
  __shared__ float lds[2*64*72];          // two 64-row x 64-col strips (padded)
  const int tid  = threadIdx.x;
  const int lane = tid & 31;
  const int w    = tid >> 5;              // wave 0..7
  const int wl   = w & 3;                 // row strip within tile
  const int g    = w >> 2;                // K-half
  int r = blockIdx.x, bi = 0;             // triangular tile decode
  while (r >= 16 - bi){ r -= 16 - bi; ++bi; }
  const int bj = bi + r;

  const int mrow = lane & 15;
  const int kb   = (lane >> 4) << 1;      // frag K base: lanes0-15->0, 16-31->2

  v8f z = {};
  v8f acc[4]; acc[0]=z; acc[1]=z; acc[2]=z; acc[3]=z;

  for (int ch = 0; ch < KCHUNKS; ++ch){
    __syncthreads();
    const int k0 = ch * 64;
    for (int t = tid; t < 2048; t += 256){        // 2 strips * 64 rows * 16 f4
      const int strip = t >> 10;
      const int rl    = (t >> 4) & 63;
      const int c4    = t & 15;
      const float* rp = resolve_row(idxArr[k0 + rl], idd, F, zrow);
      const int colb  = (strip ? bj : bi) * 64;
      const float4 v  = *(const float4*)(rp + colb + c4*4);
      float* d = &lds[strip*(64*72) + rl*72 + c4*4];
      d[0]=v.x; d[1]=v.y; d[2]=v.z; d[3]=v.w;
    }
    __syncthreads();
    const float* As = lds;
    const float* Bs = lds + 64*72;
    #pragma unroll
    for (int kk = 0; kk < 8; ++kk){
      const int r0 = g*32 + kk*4 + kb;
      v2f a;
      a.x = As[ r0     *72 + wl*16 + mrow];
      a.y = As[(r0 + 1)*72 + wl*16 + mrow];
      v2f bf[4];
      #pragma unroll
      for (int j4 = 0; j4 < 4; ++j4){
        bf[j4].x = Bs[ r0     *72 + j4*16 + mrow];
        bf[j4].y = Bs[(r0 + 1)*72 + j4*16 + mrow];
      }
      #pragma unroll
      for (int j4 = 0; j4 < 4; ++j4)
        acc[j4] = __builtin_amdgcn_wmma_f32_16x16x4_f32(
            false, a, false, bf[j4], (short)0, acc[j4], false, false);
    }
  }
  __syncthreads();
  float* xch = lds;                        // reuse staging LDS (4096 floats)
  if (g == 1){
    #pragma unroll
    for (int j4 = 0; j4 < 4; ++j4)
      #pragma unroll
      for (int v = 0; v < 8; ++v)
        xch[(((wl*4 + j4)*8 + v)*32) + lane] = acc[j4][v];
  }
  __syncthreads();
  if (g == 0){
    const int Mb  = bi*64 + wl*16 + ((lane >> 4) << 3);  // M = v + 8*(lane>=16)
    const int Nb0 = bj*64 + (lane & 15);
    #pragma unroll
    for (int j4 = 0; j4 < 4; ++j4)
      #pragma unroll
      for (int v = 0; v < 8; ++v){
        const float val = acc[j4][v] + xch[(((wl*4 + j4)*8 + v)*32) + lane];
        const int M = Mb + v;
        const int N = Nb0 + j4*16;
        G[(size_t)M*R_ + N] = val;         // tile + symmetric mirror
        G[(size_t)N*R_ + M] = val;
      }
  }
}

// ---------------------------------------------------------------------------
// K5: sigma = (G - Q * sum_c mu_c mu_c^T)/10000 + 1e-4*I  (in place)
// ---------------------------------------------------------------------------
__global__ void k_sigma(float* __restrict__ G, const float* __restrict__ mu){
  const int gid = blockIdx.x*256 + threadIdx.x;
  const int i = gid >> 10, j = gid & 1023;
  float m2 = 0.f;
  #pragma unroll
  for (int c = 0; c < C_; ++c) m2 += mu[c*R_ + i] * mu[c*R_ + j];
  float v = (G[gid] - (float)Q_ * m2) * (1.0f/10000.0f);
  if (i == j) v += 1e-4f;
  G[gid] = v;
}

// ---------------------------------------------------------------------------
// K6a: blocked Cholesky panel (64 cols, all rows below), one workgroup.
// ---------------------------------------------------------------------------
__global__ void chol_panel(float* __restrict__ A, int j0){
  __shared__ float colL[R_];
  __shared__ float sinv;
  const int tid  = threadIdx.x;
  const int jend = j0 + 64;
  for (int j = j0; j < jend; ++j){
    if (tid == 0){
      const float d = sqrtf(A[(size_t)j*R_ + j]);
      A[(size_t)j*R_ + j] = d;
      sinv = 1.0f / d;
    }
    __syncthreads();
    const float si = sinv;
    for (int i = j + 1 + tid; i < R_; i += 256){
      const float v = A[(size_t)i*R_ + j] * si;
      A[(size_t)i*R_ + j] = v;
      colL[i] = v;
    }
    __syncthreads();
    for (int i = j + 1 + tid; i < R_; i += 256){
      const float lij = colL[i];
      const int kmax = min(jend, i + 1);
      for (int k = j + 1; k < kmax; ++k)
        A[(size_t)i*R_ + k] -= lij * colL[k];
    }
    __syncthreads();
  }
}

// ---------------------------------------------------------------------------
// K6b: trailing SYRK update A[i,k] -= L_panel L_panel^T (lower triangle),
// parallel 64x64 tiles across the device.
// ---------------------------------------------------------------------------
__global__ void __launch_bounds__(256) chol_syrk(float* __restrict__ A, int j0){
  __shared__ float ldsI[64*65];
  __shared__ float ldsK[64*65];
  const int jend = j0 + 64;
  const int nb = (R_ - jend) / 64;
  int r = blockIdx.x, bi = 0;
  while (r >= nb - bi){ r -= nb - bi; ++bi; }
  const int bj = bi + r;
  const int I0 = jend + bj*64;             // rows (>= cols)
  const int K0 = jend + bi*64;             // cols
  const int tid = threadIdx.x;
  for (int t = tid; t < 1024; t += 256){
    const int rl = t >> 4, c4 = t & 15;
    const float4 vi = *(const float4*)(&A[(size_t)(I0 + rl)*R_ + j0 + c4*4]);
    const float4 vk = *(const float4*)(&A[(size_t)(K0 + rl)*R_ + j0 + c4*4]);
    float* di = &ldsI[rl*65 + c4*4];
    float* dk = &ldsK[rl*65 + c4*4];
    di[0]=vi.x; di[1]=vi.y; di[2]=vi.z; di[3]=vi.w;
    dk[0]=vk.x; dk[1]=vk.y; dk[2]=vk.z; dk[3]=vk.w;
  }
  __syncthreads();
  const int rr = tid >> 2;
  const int c0 = (tid & 3) * 16;
  float acc[16];
  #pragma unroll
  for (int c = 0; c < 16; ++c) acc[c] = 0.f;
  for (int j = 0; j < 64; ++j){
    const float a = ldsI[rr*65 + j];
    #pragma unroll
    for (int c = 0; c < 16; ++c) acc[c] += ldsK[(c0 + c)*65 + j] * a;
  }
  const int gi = I0 + rr;
  #pragma unroll
  for (int c = 0; c < 16; ++c){
    const int gk = K0 + c0 + c;
    if (gk <= gi) A[(size_t)gi*R_ + gk] -= acc[c];
  }
}

// ---------------------------------------------------------------------------
// K7: per-row ||eps||^2 + per-class argmax via packed u64 atomicMax.
// 410 MB streaming read -> the HBM-bound stage (~18us at 23.3 TB/s).
// ---------------------------------------------------------------------------
__global__ void __launch_bounds__(256) k_eps(const float* __restrict__ eps,
                                             unsigned long long* __restrict__ best){
  const int row  = blockIdx.x * 8 + (threadIdx.x >> 5);   // 0..99999
  const int lane = threadIdx.x & 31;
  const float* p = eps + (size_t)row * R_;
  float acc = 0.f;
  #pragma unroll
  for (int i = 0; i < 8; ++i){
    const float4 v = *(const float4*)(p + (i*32 + lane)*4);
    acc += v.x*v.x + v.y*v.y + v.z*v.z + v.w*v.w;
  }
  #pragma unroll
  for (int off = 16; off > 0; off >>= 1) acc += __shfl_down(acc, off, 32);
  if (lane == 0){
    const int cls = row / S_;
    const unsigned int s = (unsigned int)(row - cls*S_);
    const unsigned long long pk =
        ((unsigned long long)__float_as_uint(acc) << 32) | (unsigned long long)s;
    atomicMax(&best[cls], pk);
  }
}

// ---------------------------------------------------------------------------
// K8: vos = mu + eps_sel @ L^T (triangular matvec) ; scores_vos = vos @ W^T + b
// ---------------------------------------------------------------------------
__global__ void __launch_bounds__(256) k_vos(const unsigned long long* __restrict__ best,
      const float* __restrict__ eps, const float* __restrict__ L,
      const float* __restrict__ mu, const float* __restrict__ W,
      const float* __restrict__ bp, float* __restrict__ scoresV){
  __shared__ float epsL[R_];
  __shared__ float vosL[R_];
  __shared__ float red[256];
  const int c = blockIdx.x, t = threadIdx.x;
  const int sidx = (int)(best[c] & 0xffffffffULL);
  const float* er = eps + ((size_t)c*S_ + (size_t)sidx)*R_;
  for (int i = t; i < R_; i += 256) epsL[i] = er[i];
  __syncthreads();
  for (int i = t; i < R_; i += 256){
    const float* Lr = L + (size_t)i*R_;
    float s = 0.f;
    for (int j = 0; j <= i; ++j) s += epsL[j] * Lr[j];
    vosL[i] = mu[c*R_ + i] + s;
  }
  __syncthreads();
  for (int cc = 0; cc < C_; ++cc){
    const float* w = W + cc*R_;
    float part = 0.f;
    for (int i = t; i < R_; i += 256) part += vosL[i] * w[i];
    red[t] = part;
    __syncthreads();
    for (int off = 128; off > 0; off >>= 1){
      if (t < off) red[t] += red[t + off];
      __syncthreads();
    }
    if (t == 0) scoresV[c*C_ + cc] = red[0] + bp[cc];
    __syncthreads();
  }
}

// ---------------------------------------------------------------------------
// K9: energy -> MLP -> BCE mean * 0.1 -> out[1]
// ---------------------------------------------------------------------------
__global__ void k_dist(const float* __restrict__ scores, const float* __restrict__ scoresV,
                       const float* __restrict__ we, const float* __restrict__ W1,
                       const float* __restrict__ b1, const float* __restrict__ W2,
                       const float* __restrict__ b2, float* __restrict__ out){
  __shared__ float red[256];
  const int t = threadIdx.x;
  float local = 0.f;
  for (int row = t; row < N_ + C_; row += 256){
    const float* sr = (row < N_) ? (scores + row*C_) : (scoresV + (row - N_)*C_);
    float m = -3.402823e38f;
    #pragma unroll
    for (int c = 0; c < C_; ++c) m = fmaxf(m, sr[c]);
    float ssum = 0.f;
    #pragma unroll
    for (int c = 0; c < C_; ++c) ssum += expf(sr[c] - m) * fmaxf(we[c], 0.f);
    const float e = m + logf(ssum);
    float logit = b2[0];
    for (int k = 0; k < 512; ++k){
      const float h = fmaxf(e * W1[k] + b1[k], 0.f);
      logit += h * W2[k];
    }
    const float y = (row < N_) ? 1.f : 0.f;
    local += fmaxf(logit, 0.f) - logit*y + log1pf(expf(-fabsf(logit)));
  }
  red[t] = local;
  __syncthreads();
  for (int off = 128; off > 0; off >>= 1){
    if (t < off) red[t] += red[t + off];
    __syncthreads();
  }
  if (t == 0) out[1] = 0.1f * red[0] / (float)(N_ + C_);
}

// ---------------------------------------------------------------------------
extern "C" void kernel_launch(void* const* d_in, const int* in_sizes, int n_in,
                              void* d_out, int out_size, void* d_ws, size_t ws_size,
                              hipStream_t stream){
  (void)in_sizes; (void)n_in; (void)out_size; (void)ws_size;
  const float* F   = (const float*)d_in[0];
  const int*   lab = (const int*)  d_in[1];
  const float* idd = (const float*)d_in[2];
  // d_in[3] = id_data_counter (full queues per setup)
  const float* W   = (const float*)d_in[4];
  const float* bp  = (const float*)d_in[5];
  const float* we  = (const float*)d_in[6];
  const float* W1  = (const float*)d_in[7];
  const float* b1  = (const float*)d_in[8];
  const float* W2  = (const float*)d_in[9];
  const float* b2  = (const float*)d_in[10];
  const float* eps = (const float*)d_in[11];
  float* out = (float*)d_out;

  char* ws = (char*)d_ws;
  float*              scores  = (float*)(ws + 0);        // 20480 B
  float*              scoresV = (float*)(ws + 20480);    // 400 B
  unsigned long long* best    = (unsigned long long*)(ws + 20992); // 80 B
  int*                idxArr  = (int*)(ws + 21248);      // 40192 B -> ends 61440
  float*              zrow    = (float*)(ws + 61440);    // 4096 B
  float*              mu      = (float*)(ws + 65536);    // 40960 B
  float*              G       = (float*)(ws + 106496);   // 4 MB (G->sigma->L)

  k_init  <<<dim3(1),    dim3(256), 0, stream>>>(lab, idxArr, zrow, best);
  k_scores<<<dim3(N_),   dim3(64),  0, stream>>>(F, W, bp, scores);
  k_cls   <<<dim3(1),    dim3(512), 0, stream>>>(scores, lab, out);
  k_mu    <<<dim3(C_),   dim3(256), 0, stream>>>(idxArr, idd, F, zrow, mu);
  k_gemm  <<<dim3(136),  dim3(256), 0, stream>>>(idxArr, idd, F, zrow, G);
  k_sigma <<<dim3(4096), dim3(256), 0, stream>>>(G, mu);
  for (int s = 0; s < 16; ++s){
    chol_panel<<<dim3(1), dim3(256), 0, stream>>>(G, s*64);
    const int nb = 15 - s;
    if (nb > 0){
      const int T = nb*(nb+1)/2;
      chol_syrk<<<dim3(T), dim3(256), 0, stream>>>(G, s*64);
    }
  }
  k_eps   <<<dim3(12500), dim3(256), 0, stream>>>(eps, best);
  k_vos   <<<dim3(C_),    dim3(256), 0, stream>>>(best, eps, G, mu, W, bp, scoresV);
  k_dist  <<<dim3(1),     dim3(256), 0, stream>>>(scores, scoresV, we, W1, b1, W2, b2, out);
}